// CompGCN_24386824307090
// MI455X (gfx1250) — compile-verified
//
#include <hip/hip_runtime.h>
#include <math.h>

typedef __attribute__((ext_vector_type(2))) float v2f;
typedef __attribute__((ext_vector_type(8))) float v8f;

#define NUM_ENT    50000
#define DIM        200
#define NUM_EDGES  1000000
#define HALF_EDGES 500000
#define NREL2      500        // 2*NUM_REL
#define NKEYS      1000       // (dir, etype) groups
#define BATCH      2048
#define NTILES_N   13         // ceil(200/16)
#define NSUB       4          // decoder: 16x64 output per wave
#define GSPLIT     8          // blocks per edge-group

// ---------------------------------------------------------------------------
// Sort phase: counting sort of edges by key = dir*500 + etype.
// ---------------------------------------------------------------------------
__global__ void hist_kernel(const int* __restrict__ etype, int* __restrict__ hist)
{
  int e = blockIdx.x * blockDim.x + threadIdx.x;
  if (e >= NUM_EDGES) return;
  int key = (e < HALF_EDGES ? 0 : NREL2) + etype[e];
  atomicAdd(hist + key, 1);
}

__global__ void scan_kernel(const int* __restrict__ hist,
                            int* __restrict__ binStart, int* __restrict__ cursor)
{
  if (threadIdx.x == 0 && blockIdx.x == 0) {
    int s = 0;
    for (int i = 0; i < NKEYS; ++i) { binStart[i] = s; cursor[i] = s; s += hist[i]; }
    binStart[NKEYS] = s;
  }
}

__global__ void scatter_kernel(const int* __restrict__ etype,
                               int* __restrict__ cursor, int* __restrict__ order)
{
  int e = blockIdx.x * blockDim.x + threadIdx.x;
  if (e >= NUM_EDGES) return;
  int key = (e < HALF_EDGES ? 0 : NREL2) + etype[e];
  int pos = atomicAdd(cursor + key, 1);
  order[pos] = e;
}

// ---------------------------------------------------------------------------
// Edge ccorr as WMMA GEMM: within group g all edges share B = circ(rel_emb[r]).
// data = A_gathered(16xK) x B(KxN), B[i][n] = b[(i+n)%200] from duplicated LDS.
// Result rows scaled by norm[e] and atomically scattered into P[dir][dst[e]].
// ---------------------------------------------------------------------------
__global__ __launch_bounds__(128) void edge_gemm_kernel(
    const float* __restrict__ ent_emb, const float* __restrict__ rel_emb,
    const float* __restrict__ norm, const int* __restrict__ src,
    const int* __restrict__ dst, const int* __restrict__ order,
    const int* __restrict__ binStart, float* __restrict__ P)
{
  __shared__ float bdup[2*DIM + 16];
  const int g    = blockIdx.x;                 // 0..999
  const int base = binStart[g];
  const int cnt  = binStart[g + 1] - base;
  {
    const float* brow = rel_emb + (long)(g % NREL2) * DIM;
    for (int i = threadIdx.x; i < DIM; i += 128) {
      float v = brow[i];
      bdup[i] = v; bdup[i + DIM] = v;
    }
    for (int i = 2*DIM + threadIdx.x; i < 2*DIM + 16; i += 128) bdup[i] = 0.f;
  }
  __syncthreads();
  if (cnt == 0) return;

  const int lane = threadIdx.x & 31;
  const int wid  = threadIdx.x >> 5;
  const int l15  = lane & 15;
  const int kk   = (lane >> 4) << 1;           // A/B K-pair offset per lane half
  const int tiles = ((cnt + 15) >> 4) * NTILES_N;
  float* Pd = P + (g < NREL2 ? 0l : (long)NUM_ENT * DIM);

  for (int t = blockIdx.y * 4 + wid; t < tiles; t += GSPLIT * 4) {
    const int tm = t / NTILES_N;
    const int tn = t % NTILES_N;
    const int n0 = tn * 16;

    int mA = tm * 16 + l15; if (mA >= cnt) mA = cnt - 1;   // clamp tail rows
    const int eA = order[base + mA];
    const float* arow = ent_emb + (long)src[eA] * DIM;

    v8f acc = {};
    const int bofs = n0 + l15 + kk;            // bdup[(k+kk) + n], n = n0+l15
    #pragma unroll 2
    for (int k = 0; k < DIM; k += 4) {
      v2f a = *(const v2f*)(arow + k + kk);    // global_load_b64
      v2f b; b.x = bdup[k + bofs]; b.y = bdup[k + bofs + 1];
      acc = __builtin_amdgcn_wmma_f32_16x16x4_f32(false, a, false, b,
                                                  (short)0, acc, false, false);
    }

    // Scatter: D layout: lane holds rows m = mbase+vv, col n = n0 + l%16.
    const int n = n0 + l15;
    const int mbase = tm * 16 + ((lane >> 4) << 3);
    if (n < DIM) {
      if (tm * 16 + 16 <= cnt) {
        // fast path (full tile): branch-free, loads batch into clauses
        int e8[8];
        #pragma unroll
        for (int vv = 0; vv < 8; ++vv) e8[vv] = order[base + mbase + vv];
        float nv8[8];
        long  ro8[8];
        #pragma unroll
        for (int vv = 0; vv < 8; ++vv) {
          nv8[vv] = norm[e8[vv]];
          ro8[vv] = (long)dst[e8[vv]] * DIM;
        }
        #pragma unroll
        for (int vv = 0; vv < 8; ++vv)
          atomicAdd(Pd + ro8[vv] + n, acc[vv] * nv8[vv]);
      } else {
        // tail tile (rare): guarded
        #pragma unroll
        for (int vv = 0; vv < 8; ++vv) {
          int m = mbase + vv;
          if (m < cnt) {
            int e = order[base + m];
            atomicAdd(Pd + (long)dst[e] * DIM + n, acc[vv] * norm[e]);
          }
        }
      }
    }
  }
}

// ---------------------------------------------------------------------------
// Self-loop ccorr as WMMA GEMM: loop_pre = ent_emb @ circ(loop_rel)
// ---------------------------------------------------------------------------
__global__ __launch_bounds__(128) void loop_gemm_kernel(
    const float* __restrict__ ent_emb, const float* __restrict__ loop_rel,
    float* __restrict__ loop_pre)
{
  __shared__ float bdup[2*DIM + 16];
  for (int i = threadIdx.x; i < DIM; i += 128) {
    float v = loop_rel[i];
    bdup[i] = v; bdup[i + DIM] = v;
  }
  for (int i = 2*DIM + threadIdx.x; i < 2*DIM + 16; i += 128) bdup[i] = 0.f;
  __syncthreads();

  const int lane = threadIdx.x & 31;
  const int wid  = threadIdx.x >> 5;
  const int t = blockIdx.x * 4 + wid;
  const int tiles = (NUM_ENT / 16) * NTILES_N;
  if (t >= tiles) return;
  const int tm = t / NTILES_N, tn = t % NTILES_N;
  const int n0 = tn * 16;
  const int l15 = lane & 15;
  const int kk  = (lane >> 4) << 1;
  const float* arow = ent_emb + (long)(tm * 16 + l15) * DIM;

  v8f acc = {};
  const int bofs = n0 + l15 + kk;
  #pragma unroll 2
  for (int k = 0; k < DIM; k += 4) {
    v2f a = *(const v2f*)(arow + k + kk);
    v2f b; b.x = bdup[k + bofs]; b.y = bdup[k + bofs + 1];
    acc = __builtin_amdgcn_wmma_f32_16x16x4_f32(false, a, false, b,
                                                (short)0, acc, false, false);
  }
  const int n = n0 + l15;
  if (n < DIM) {
    #pragma unroll
    for (int vv = 0; vv < 8; ++vv) {
      int m = tm * 16 + vv + ((lane >> 4) << 3);
      loop_pre[(long)m * DIM + n] = acc[vv];
    }
  }
}

// ---------------------------------------------------------------------------
// h = (Pin@in_w + Pout@out_w + loop_pre@loop_w)/3  (fp32 WMMA) + fused stats.
// B column index clamped (no divergence in the k loop); tail discarded at store.
// ---------------------------------------------------------------------------
__global__ __launch_bounds__(128) void h_gemm_kernel(
    const float* __restrict__ Pin, const float* __restrict__ Pout,
    const float* __restrict__ loop_pre,
    const float* __restrict__ in_w, const float* __restrict__ out_w,
    const float* __restrict__ loop_w,
    float* __restrict__ h, float* __restrict__ colsum, float* __restrict__ colsq)
{
  const int lane = threadIdx.x & 31;
  const int wid  = threadIdx.x >> 5;
  const int t = blockIdx.x * 4 + wid;
  const int tiles = (NUM_ENT / 16) * NTILES_N;
  if (t >= tiles) return;
  const int mt = t / NTILES_N, nt = t % NTILES_N;
  const int m0 = mt * 16, n0 = nt * 16;
  const int mrow = m0 + (lane & 15);
  const int kk = (lane >> 4) << 1;
  const int n = n0 + (lane & 15);
  const int ncl = (n < DIM) ? n : (DIM - 1);   // clamped: no EXEC churn in loop

  v8f acc = {};
  const float* Ss[3] = {Pin, Pout, loop_pre};
  const float* Ws[3] = {in_w, out_w, loop_w};
  #pragma unroll
  for (int p = 0; p < 3; ++p) {
    const float* S = Ss[p];
    const float* W = Ws[p] + ncl;
    for (int k = 0; k < DIM; k += 4) {
      v2f a = *(const v2f*)(S + (long)mrow * DIM + k + kk);
      v2f b; b.x = W[(k + kk) * DIM]; b.y = W[(k + kk + 1) * DIM];
      acc = __builtin_amdgcn_wmma_f32_16x16x4_f32(false, a, false, b,
                                                  (short)0, acc, false, false);
    }
  }
  const float inv3 = 1.f / 3.f;
  float cs = 0.f, cq = 0.f;
  #pragma unroll
  for (int vv = 0; vv < 8; ++vv) {
    float hv = acc[vv] * inv3;
    int mr = m0 + vv + ((lane >> 4) << 3);
    if (n < DIM) h[(long)mr * DIM + n] = hv;
    cs += hv; cq += hv * hv;
  }
  if (n < DIM) { atomicAdd(colsum + n, cs); atomicAdd(colsq + n, cq); }
}

// Finalize batch-norm stats
__global__ void stats_kernel(const float* __restrict__ colsum,
                             const float* __restrict__ colsq,
                             float* __restrict__ mean, float* __restrict__ rstd)
{
  int nidx = threadIdx.x + blockIdx.x * blockDim.x;
  if (nidx < DIM) {
    float m = colsum[nidx] * (1.f / NUM_ENT);
    float v = colsq[nidx] * (1.f / NUM_ENT) - m * m;
    mean[nidx] = m;
    rstd[nidx] = rsqrtf(v + 1e-5f);
  }
}

// r = rel_emb @ w_rel (tiny: 20 MMACs total)
__global__ void relw_kernel(const float* __restrict__ rel_emb,
                            const float* __restrict__ w_rel, float* __restrict__ r)
{
  int idx = blockIdx.x * blockDim.x + threadIdx.x;
  if (idx >= NREL2 * DIM) return;
  int i = idx / DIM, n = idx % DIM;
  float s = 0.f;
  for (int k = 0; k < DIM; ++k) s = fmaf(rel_emb[i * DIM + k], w_rel[k * DIM + n], s);
  r[idx] = s;
}

// obj[t] = tanh((h[head]-mean)*rstd) * r[rela]
__global__ void obj_kernel(const float* __restrict__ h, const float* __restrict__ mean,
                           const float* __restrict__ rstd, const float* __restrict__ r,
                           const int* __restrict__ triples, float* __restrict__ obj)
{
  int idx = blockIdx.x * blockDim.x + threadIdx.x;
  if (idx >= BATCH * DIM) return;
  int t = idx / DIM, n = idx % DIM;
  int head = triples[t * 3 + 0];
  int rela = triples[t * 3 + 1];
  float x = tanhf((h[(long)head * DIM + n] - mean[n]) * rstd[n]);
  obj[idx] = x * r[(long)rela * DIM + n];
}

// ---------------------------------------------------------------------------
// Decoder: sigmoid(obj @ emb_ent_w.T + bias), fp32 WMMA, 16x64 per wave.
// n0 clamped so every load/store is in range (overlap rewrites identical data).
// ---------------------------------------------------------------------------
__global__ __launch_bounds__(128) void dec_gemm_kernel(
    const float* __restrict__ obj, const float* __restrict__ emb_ent_w,
    const float* __restrict__ dec_bias, float* __restrict__ out)
{
  const int lane = threadIdx.x & 31;
  const int wid  = threadIdx.x >> 5;
  const int mtiles  = BATCH / 16;                         // 128
  const int nsupers = (NUM_ENT / 16 + NSUB - 1) / NSUB;   // 782
  const int t = blockIdx.x * 4 + wid;
  if (t >= mtiles * nsupers) return;
  const int mt = t / nsupers, ns = t % nsupers;
  const int m0 = mt * 16;
  int n0 = ns * (16 * NSUB);
  if (n0 > NUM_ENT - 16 * NSUB) n0 = NUM_ENT - 16 * NSUB; // clamp: all-valid tile
  const int mrow = m0 + (lane & 15);
  const int kk = (lane >> 4) << 1;
  const int l15 = lane & 15;

  v8f acc[NSUB];
  v8f z = {};
  #pragma unroll
  for (int c = 0; c < NSUB; ++c) acc[c] = z;

  for (int k = 0; k < DIM; k += 4) {
    v2f a = *(const v2f*)(obj + (long)mrow * DIM + k + kk);
    #pragma unroll
    for (int c = 0; c < NSUB; ++c) {
      const long n = n0 + c * 16 + l15;
      v2f b = *(const v2f*)(emb_ent_w + n * DIM + k + kk); // transpose is free
      acc[c] = __builtin_amdgcn_wmma_f32_16x16x4_f32(false, a, false, b,
                                                     (short)0, acc[c], false, false);
    }
  }
  #pragma unroll
  for (int c = 0; c < NSUB; ++c) {
    const long n = n0 + c * 16 + l15;
    float bias = dec_bias[n];
    #pragma unroll
    for (int vv = 0; vv < 8; ++vv) {
      int mr = m0 + vv + ((lane >> 4) << 3);
      float s = acc[c][vv] + bias;
      out[(long)mr * NUM_ENT + n] = 1.f / (1.f + __expf(-s));
    }
  }
}

// ---------------------------------------------------------------------------
extern "C" void kernel_launch(void* const* d_in, const int* in_sizes, int n_in,
                              void* d_out, int out_size, void* d_ws, size_t ws_size,
                              hipStream_t stream)
{
  const float* ent_emb   = (const float*)d_in[0];
  const float* rel_emb   = (const float*)d_in[1];
  const float* in_w      = (const float*)d_in[2];
  const float* out_w     = (const float*)d_in[3];
  const float* loop_w    = (const float*)d_in[4];
  const float* w_rel     = (const float*)d_in[5];
  const float* loop_rel  = (const float*)d_in[6];
  const float* emb_ent_w = (const float*)d_in[7];
  const float* dec_bias  = (const float*)d_in[8];
  const float* norm      = (const float*)d_in[9];
  const int* src     = (const int*)d_in[10];
  const int* dst     = (const int*)d_in[11];
  const int* etype   = (const int*)d_in[12];
  const int* triples = (const int*)d_in[13];
  float* out = (float*)d_out;

  float* ws = (float*)d_ws;
  const long PSZ = (long)NUM_ENT * DIM;          // 10,000,000 floats
  float* Pin      = ws;                           // [2][NUM_ENT][DIM]
  float* loop_pre = ws + 2 * PSZ;
  float* h        = ws + 3 * PSZ;
  float* colsum   = ws + 4 * PSZ;
  float* colsq    = colsum + DIM;
  float* mean     = colsq + DIM;
  float* rstd     = mean + DIM;
  float* r        = rstd + DIM;                   // [500][200]
  float* obj      = r + (long)NREL2 * DIM;        // [2048][200]
  int* hist     = (int*)(obj + (long)BATCH * DIM);
  int* binStart = hist + NKEYS;                   // NKEYS+1 entries
  int* cursor   = binStart + NKEYS + 1;
  int* order    = cursor + NKEYS;                 // [NUM_EDGES]
  (void)ws_size; (void)in_sizes; (void)n_in; (void)out_size;

  hipMemsetAsync(Pin, 0, (size_t)(2 * PSZ) * sizeof(float), stream);
  hipMemsetAsync(colsum, 0, (size_t)(2 * DIM) * sizeof(float), stream);
  hipMemsetAsync(hist, 0, (size_t)NKEYS * sizeof(int), stream);

  // counting sort of edges by (dir, etype)
  hist_kernel<<<(NUM_EDGES + 255) / 256, 256, 0, stream>>>(etype, hist);
  scan_kernel<<<1, 32, 0, stream>>>(hist, binStart, cursor);
  scatter_kernel<<<(NUM_EDGES + 255) / 256, 256, 0, stream>>>(etype, cursor, order);

  // per-group gathered WMMA GEMM + scatter into P
  edge_gemm_kernel<<<dim3(NKEYS, GSPLIT), 128, 0, stream>>>(
      ent_emb, rel_emb, norm, src, dst, order, binStart, Pin);

  // self-loop ccorr GEMM
  {
    int tiles = (NUM_ENT / 16) * NTILES_N;
    loop_gemm_kernel<<<(tiles + 3) / 4, 128, 0, stream>>>(ent_emb, loop_rel, loop_pre);
  }
  // entity update GEMM + batch-norm stats
  {
    int tiles = (NUM_ENT / 16) * NTILES_N;
    h_gemm_kernel<<<(tiles + 3) / 4, 128, 0, stream>>>(
        Pin, Pin + PSZ, loop_pre, in_w, out_w, loop_w, h, colsum, colsq);
  }
  stats_kernel<<<1, 256, 0, stream>>>(colsum, colsq, mean, rstd);
  relw_kernel<<<(NREL2 * DIM + 255) / 256, 256, 0, stream>>>(rel_emb, w_rel, r);
  obj_kernel<<<(BATCH * DIM + 255) / 256, 256, 0, stream>>>(
      h, mean, rstd, r, triples, obj);
  // decoder GEMM
  {
    int t = (BATCH / 16) * ((NUM_ENT / 16 + NSUB - 1) / NSUB);
    dec_gemm_kernel<<<(t + 3) / 4, 128, 0, stream>>>(obj, emb_ent_w, dec_bias, out);
  }
}